// OuterPosBow_68616397521347
// MI455X (gfx1250) — compile-verified
//
#include <hip/hip_runtime.h>

// Problem constants (match reference)
#define B_      64
#define NW_     256
#define L_      32
#define NCH     64          // char vocab
#define E3_     256         // per-slot embedding dim
#define ROWLEN  770         // output row length (2 pad + 3*E3)

#define WT_STRIDE  257      // 257 % 64 == 1 -> conflict-free transpose writes
#define CNT_STRIDE 65

typedef __attribute__((ext_vector_type(2))) float v2f;
typedef __attribute__((ext_vector_type(8))) float v8f;

__global__ __launch_bounds__(256) void OuterPosBow_68616397521347_kernel(
    const int* __restrict__ sntcs, const float* __restrict__ W,
    float* __restrict__ out)
{
    extern __shared__ char smem_raw[];
    float* sWt    = (float*)smem_raw;                 // [NCH][WT_STRIDE]  Wt[c][e] = W[e][c]
    float* sCnt   = sWt + NCH * WT_STRIDE;            // [16][CNT_STRIDE]  f32 char counts
    int*   sFirst = (int*)(sCnt + 16 * CNT_STRIDE);   // [16] first-char index per word
    int*   sEnds  = sFirst + 16;                      // [16] ends-char index per word

    const int t  = threadIdx.x;                       // 0..255
    const int m0 = blockIdx.x * 16;                   // first word of this M-tile

    // ---- Stage W (E3 x NCH, row-major) transposed into LDS ----
    for (int i = t; i < E3_ * NCH; i += 256) {
        const float v = W[i];
        const int e = i >> 6;      // 0..255
        const int c = i & 63;      // 0..63
        sWt[c * WT_STRIDE + e] = v;
    }

    // ---- Per-word metadata + char-count rows (A matrix of the GEMM) ----
    if (t < 16) {
        const int* sp = sntcs + (size_t)(m0 + t) * L_;
        int best = -1, bidx = 0, c0 = 0;
        #pragma unroll
        for (int j = 0; j < L_; ++j) {
            const int v = sp[j];
            if (j == 0) c0 = v;
            if (v > best) { best = v; bidx = j; }   // first occurrence of max
        }
        int wl = bidx - 1; if (wl < 0) wl = 0;       // relu(argmax - 1), in [0,30]
        const int ends = sp[wl];
        sFirst[t] = (wl == 0) ? 0 : c0;              // scatter zeroed position 0?
        sEnds[t]  = ends;
        float* cr = sCnt + t * CNT_STRIDE;
        #pragma unroll
        for (int c = 0; c < NCH; ++c) cr[c] = 0.0f;
        #pragma unroll
        for (int j = 1; j <= L_ - 2; ++j) {          // interior positions 1..30
            const int c = (j == wl) ? 0 : sp[j];     // scatter-zero at wl
            cr[c] += 1.0f;
        }
    }
    __syncthreads();

    // ---- bow = counts[16x64] @ Wt[64x256] via V_WMMA_F32_16X16X4_F32 ----
    const int lane = t & 31;
    const int wv   = t >> 5;          // wave 0..7 owns two N-tiles
    const int l16  = lane & 15;
    const int hi   = lane >> 4;       // 0: K+0 pair, 1: K+2 pair (ISA fragment layout)
    const int n0a  = (wv * 2 + 0) * 16;
    const int n0b  = (wv * 2 + 1) * 16;

    v8f acc0 = {};
    v8f acc1 = {};
    #pragma unroll
    for (int step = 0; step < NCH / 4; ++step) {
        const int ka = step * 4 + 2 * hi;            // K held in VGPR0 for this lane
        v2f a, b0, b1;
        a.x  = sCnt[l16 * CNT_STRIDE + ka];          // A: row M=l16
        a.y  = sCnt[l16 * CNT_STRIDE + ka + 1];
        b0.x = sWt[ ka      * WT_STRIDE + n0a + l16]; // B: row K, col N=n0+l16
        b0.y = sWt[(ka + 1) * WT_STRIDE + n0a + l16];
        b1.x = sWt[ ka      * WT_STRIDE + n0b + l16];
        b1.y = sWt[(ka + 1) * WT_STRIDE + n0b + l16];
        acc0 = __builtin_amdgcn_wmma_f32_16x16x4_f32(false, a, false, b0,
                                                     (short)0, acc0, false, false);
        acc1 = __builtin_amdgcn_wmma_f32_16x16x4_f32(false, a, false, b1,
                                                     (short)0, acc1, false, false);
    }

    // ---- store bow block: D layout VGPR r -> row (r + 8*hi), col n0+l16 ----
    #pragma unroll
    for (int r = 0; r < 8; ++r) {
        const int m = r + 8 * hi;
        const size_t base = (size_t)(m0 + m) * ROWLEN + 2 + E3_;   // col 258
        out[base + n0a + l16] = acc0[r];
        out[base + n0b + l16] = acc1[r];
    }

    // ---- first-char block (cols 2..257), ends block (cols 514..769) ----
    for (int m = 0; m < 16; ++m) {
        const size_t base = (size_t)(m0 + m) * ROWLEN;
        const int fc = sFirst[m];
        const int ec = sEnds[m];
        out[base + 2 + t]            = sWt[fc * WT_STRIDE + t];
        out[base + 2 + 2 * E3_ + t]  = sWt[ec * WT_STRIDE + t];   // col 514 + t
    }

    // ---- left zero pad (cols 0..1) ----
    if (t < 32) {
        out[(size_t)(m0 + (t >> 1)) * ROWLEN + (t & 1)] = 0.0f;
    }
}

extern "C" void kernel_launch(void* const* d_in, const int* in_sizes, int n_in,
                              void* d_out, int out_size, void* d_ws, size_t ws_size,
                              hipStream_t stream) {
    (void)in_sizes; (void)n_in; (void)out_size; (void)d_ws; (void)ws_size;
    const int*   sntcs = (const int*)d_in[0];   // [B, NW, L] int32
    const float* W     = (const float*)d_in[1]; // [E3, NCH] float32
    float*       out   = (float*)d_out;         // [B, NW, 770] float32

    const int nTiles = (B_ * NW_) / 16;         // 1024 blocks
    const size_t smem = (size_t)(NCH * WT_STRIDE + 16 * CNT_STRIDE) * sizeof(float)
                      + 32 * sizeof(int);       // 70,080 bytes dynamic LDS
    OuterPosBow_68616397521347_kernel<<<dim3(nTiles), dim3(256), smem, stream>>>(
        sntcs, W, out);
}